// FrangiFilter_27805618274932
// MI455X (gfx1250) — compile-verified
//
#include <hip/hip_runtime.h>
#include <math.h>

// MI455X / gfx1250, wave32. Vertical convolutions run on the matrix pipe via
// V_WMMA_F32_16X16X4_F32 (banded-weight-matrix x input-rows), full f32 precision.
// Each wave processes a 16x32 output strip (two 16x16 WMMA tiles) so the banded
// A-fragments are reused across 2 tiles -> 6 WMMAs per K-chunk.

typedef __attribute__((ext_vector_type(2))) float v2f;
typedef __attribute__((ext_vector_type(8))) float v8f;

#define IMG_H 2048
#define IMG_W 2048
#define PLANE (IMG_H * IMG_W)
#define NSIG 5
#define WSTRIDE 240   // per-sigma weight slot: g[80], g1[80], g2[80]
#define RMAX 36

static __device__ __forceinline__ int refl(int i, int n) {
    // np.pad(mode='reflect'): mirror without repeating the edge sample
    i = (i < 0) ? -i : i;
    i = (i >= n) ? (2 * n - 2 - i) : i;
    return i;
}

// ---------------------------------------------------------------------------
// Zero the ssq-max accumulator (d_ws is poisoned before timing).
__global__ void frangi_init_stats(unsigned int* ssqmax) {
    if (threadIdx.x == 0) *ssqmax = 0u;
}

// ---------------------------------------------------------------------------
// Build sampled Gaussian / 1st / 2nd derivative kernels for all sigmas
// (double precision, matching the reference's float64 kernel construction).
__global__ void frangi_weights(float* __restrict__ w) {
    __shared__ double red[128];
    const double sig[NSIG] = {1.0, 3.0, 5.0, 7.0, 9.0};
    for (int s = 0; s < NSIG; ++s) {
        const double sigma = sig[s];
        const int r = (int)(4.0 * sigma + 0.5);
        const int L = 2 * r + 1;
        const double s2 = sigma * sigma;
        const int i = threadIdx.x;
        const double x = (double)(i - r);
        double gv = 0.0;
        if (i < L) gv = exp(-x * x / (2.0 * s2));
        red[i] = gv;
        __syncthreads();
        for (int st = 64; st > 0; st >>= 1) {
            if (i < st) red[i] += red[i + st];
            __syncthreads();
        }
        const double sum = red[0];
        __syncthreads();
        if (i < L) {
            const double g = gv / sum;
            float* base = w + s * WSTRIDE;
            base[i]        = (float)g;                                   // g
            base[80 + i]   = (float)((-x / s2) * g);                     // g1
            base[160 + i]  = (float)(((x * x - s2) / (s2 * s2)) * g);    // g2
        }
        __syncthreads();
    }
}

// ---------------------------------------------------------------------------
// Grayscale (torchvision weights), negated for black_ridges. float4 vectorized.
__global__ void frangi_gray(const float* __restrict__ img, float* __restrict__ gray) {
    const int idx = blockIdx.x * blockDim.x + threadIdx.x;  // over PLANE/4
    const float4 R = ((const float4*)img)[idx];
    const float4 G = ((const float4*)(img + PLANE))[idx];
    const float4 B = ((const float4*)(img + 2 * (size_t)PLANE))[idx];
    float4 o;
    o.x = -(0.2989f * R.x + 0.587f * G.x + 0.114f * B.x);
    o.y = -(0.2989f * R.y + 0.587f * G.y + 0.114f * B.y);
    o.z = -(0.2989f * R.z + 0.587f * G.z + 0.114f * B.z);
    o.w = -(0.2989f * R.w + 0.587f * G.w + 0.114f * B.w);
    ((float4*)gray)[idx] = o;
}

// ---------------------------------------------------------------------------
// Horizontal (x-axis) correlation with g, g1, g2 simultaneously, reflect pad.
// One block = one 256-pixel row segment; input segment cached in LDS once,
// reused by all three kernels.
__global__ void frangi_rowconv(const float* __restrict__ gray,
                               const float* __restrict__ wsig,
                               float* __restrict__ t0, float* __restrict__ t1,
                               float* __restrict__ t2, int r, int L) {
    __shared__ float s_in[256 + 2 * RMAX];
    __shared__ float s_w[WSTRIDE];
    const int tid = threadIdx.x;
    const int y = blockIdx.y;
    const int x0 = blockIdx.x * 256;
    for (int i = tid; i < WSTRIDE; i += 256)
        s_w[i] = ((i % 80) < L) ? wsig[i] : 0.f;
    const float* row = gray + (size_t)y * IMG_W;
    const int ext = 256 + 2 * r;
    for (int i = tid; i < ext; i += 256)
        s_in[i] = row[refl(x0 + i - r, IMG_W)];
    __syncthreads();
    float a0 = 0.f, a1 = 0.f, a2 = 0.f;
    for (int t = 0; t < L; ++t) {
        const float v = s_in[tid + t];
        a0 = fmaf(s_w[t], v, a0);         // g  along x
        a1 = fmaf(s_w[80 + t], v, a1);    // g1 along x
        a2 = fmaf(s_w[160 + t], v, a2);   // g2 along x
    }
    const size_t o = (size_t)y * IMG_W + x0 + tid;
    t0[o] = a0; t1[o] = a1; t2[o] = a2;
}

// ---------------------------------------------------------------------------
// Vertical (y-axis) correlation via V_WMMA_F32_16X16X4_F32 + fused Hessian
// eigen-analysis / vesselness epilogue.
//   hrr = (g2 along y) of t0 ; hcc = (g along y) of t2 ; hrc = (g1 along y) of t1
// Banded matmul: D[m][n] = sum_k W[m][k] * B[k][n], W[m][k] = w[k-m] (0 outside),
// B[k][n] = tmp[refl(y0+k-r)][x0+n], K chunked by 4, accumulated through C.
// A-frag layout (f32 16x4): lane m=lane&15; VGPR0/1 hold K = kc+2*(lane>>4) +0/+1.
// B-frag layout mirrors A (rows K striped the same way); C/D: elem j of v8f is
// row M = j + 8*(lane>>4), column N = lane&15.
// Each wave covers x0..x0+31 (two tiles) to reuse A-fragments across 6 WMMAs.
// mode: 0 = reduce max(ssq) only (sigma=1 gamma pass)
//       1 = compute vals and STORE to out (first scale; out is poisoned)
//       2 = compute vals and out = max(out, vals)
__global__ void frangi_colconv_wmma(const float* __restrict__ t0,
                                    const float* __restrict__ t1,
                                    const float* __restrict__ t2,
                                    const float* __restrict__ wsig,
                                    float* __restrict__ out,
                                    unsigned int* __restrict__ ssqmax,
                                    const float* __restrict__ inv2g2p,
                                    float s2, int r, int L, int mode) {
    __shared__ float s_w[WSTRIDE];
    __shared__ unsigned int s_max;
    const int tid = threadIdx.x;
    if (tid == 0) s_max = 0u;
    for (int i = tid; i < WSTRIDE; i += 256)
        s_w[i] = ((i % 80) < L) ? wsig[i] : 0.f;
    __syncthreads();

    const int lane = tid & 31;
    const int wv   = tid >> 5;           // 8 waves per block
    const int m    = lane & 15;          // A-row / B,C column within tile
    const int hi   = lane >> 4;          // lane-half selects K(+2) / M(+8)
    const int x0   = (blockIdx.x * 8 + wv) * 32;   // gridDim.x = 8 -> 64 strips
    const int y0   = blockIdx.y * 16;              // gridDim.y = 128

    v8f hrr0 = {}; v8f hcc0 = {}; v8f hrc0 = {};
    v8f hrr1 = {}; v8f hcc1 = {}; v8f hrc1 = {};
    const int nch  = (L + 18) >> 2;
    const int colA = x0 + m;
    const int colB = x0 + 16 + m;
    // Border tiles (top/bottom ~3 tile rows) need reflect indexing; interior skips it.
    const bool interior = (y0 - r >= 0) && (y0 + 4 * nch - 1 - r < IMG_H);

    for (int c = 0; c < nch; ++c) {
        const int kc = 4 * c + 2 * hi;       // this lane-half's first K index
        const int i0 = kc - m, i1 = i0 + 1;
        const bool u0 = (i0 >= 0) & (i0 < L);
        const bool u1 = (i1 >= 0) & (i1 < L);
        v2f ag, ag1, ag2;
        ag.x  = u0 ? s_w[i0]       : 0.f;   ag.y  = u1 ? s_w[i1]       : 0.f;
        ag1.x = u0 ? s_w[80 + i0]  : 0.f;   ag1.y = u1 ? s_w[80 + i1]  : 0.f;
        ag2.x = u0 ? s_w[160 + i0] : 0.f;   ag2.y = u1 ? s_w[160 + i1] : 0.f;

        int yA = y0 + kc - r;
        int yB = yA + 1;
        if (!interior) { yA = refl(yA, IMG_H); yB = refl(yB, IMG_H); }
        const size_t oA = (size_t)yA * IMG_W;
        const size_t oB = (size_t)yB * IMG_W;

        v2f b0a, b1a, b2a, b0b, b1b, b2b;
        b0a.x = t0[oA + colA]; b0a.y = t0[oB + colA];
        b1a.x = t1[oA + colA]; b1a.y = t1[oB + colA];
        b2a.x = t2[oA + colA]; b2a.y = t2[oB + colA];
        b0b.x = t0[oA + colB]; b0b.y = t0[oB + colB];
        b1b.x = t1[oA + colB]; b1b.y = t1[oB + colB];
        b2b.x = t2[oA + colB]; b2b.y = t2[oB + colB];

        hrr0 = __builtin_amdgcn_wmma_f32_16x16x4_f32(false, ag2, false, b0a,
                                                     (short)0, hrr0, false, false);
        hrr1 = __builtin_amdgcn_wmma_f32_16x16x4_f32(false, ag2, false, b0b,
                                                     (short)0, hrr1, false, false);
        hcc0 = __builtin_amdgcn_wmma_f32_16x16x4_f32(false, ag,  false, b2a,
                                                     (short)0, hcc0, false, false);
        hcc1 = __builtin_amdgcn_wmma_f32_16x16x4_f32(false, ag,  false, b2b,
                                                     (short)0, hcc1, false, false);
        hrc0 = __builtin_amdgcn_wmma_f32_16x16x4_f32(false, ag1, false, b1a,
                                                     (short)0, hrc0, false, false);
        hrc1 = __builtin_amdgcn_wmma_f32_16x16x4_f32(false, ag1, false, b1b,
                                                     (short)0, hrc1, false, false);
    }

    const float inv2g2 = (mode == 0) ? 0.f : *inv2g2p;
    float lmax = 0.f;
#pragma unroll
    for (int t = 0; t < 2; ++t) {
        const v8f hrr = (t == 0) ? hrr0 : hrr1;
        const v8f hcc = (t == 0) ? hcc0 : hcc1;
        const v8f hrc = (t == 0) ? hrc0 : hrc1;
        const int col = (t == 0) ? colA : colB;
#pragma unroll
        for (int j = 0; j < 8; ++j) {
            const float a = s2 * hrr[j];
            const float d = s2 * hcc[j];
            const float b = s2 * hrc[j];
            const float mid  = 0.5f * (a + d);
            const float dif  = a - d;
            const float disc = sqrtf(0.25f * dif * dif + b * b);
            const float elo = mid - disc, ehi = mid + disc;
            const bool swp = fabsf(elo) > fabsf(ehi);
            const float lam1 = swp ? ehi : elo;
            const float lam2 = swp ? elo : ehi;
            const float lam2c = fmaxf(lam2, 1e-10f);
            const float rb = lam1 / lam2c;
            const float rb2 = rb * rb;
            const float ssq = lam1 * lam1 + lam2 * lam2;
            if (mode == 0) {
                lmax = fmaxf(lmax, ssq);
            } else {
                const float v = __expf(-2.f * rb2) * (1.f - __expf(-ssq * inv2g2));
                const size_t o = (size_t)(y0 + 8 * hi + j) * IMG_W + col;
                out[o] = (mode == 1) ? v : fmaxf(out[o], v);
            }
        }
    }
    if (mode == 0) {
        atomicMax(&s_max, __float_as_uint(lmax));   // ssq >= 0: bits order as uint
        __syncthreads();
        if (tid == 0) atomicMax(ssqmax, s_max);
    }
}

// ---------------------------------------------------------------------------
// gamma = 0.5*sqrt(max_ssq)  =>  1/(2 gamma^2) = 2/max_ssq  (0.5 if max==0)
__global__ void frangi_finalize(const unsigned int* __restrict__ ssqmax,
                                float* __restrict__ inv2g2) {
    const float mx = __uint_as_float(*ssqmax);
    *inv2g2 = (mx == 0.f) ? 0.5f : (2.0f / mx);
}

// ---------------------------------------------------------------------------
extern "C" void kernel_launch(void* const* d_in, const int* in_sizes, int n_in,
                              void* d_out, int out_size, void* d_ws, size_t ws_size,
                              hipStream_t stream) {
    const float* img = (const float*)d_in[0];
    float* out = (float*)d_out;
    float* ws  = (float*)d_ws;

    float* gray = ws;
    float* t0 = ws + (size_t)PLANE;          // g  along x
    float* t1 = ws + 2 * (size_t)PLANE;      // g1 along x
    float* t2 = ws + 3 * (size_t)PLANE;      // g2 along x
    float* wts = ws + 4 * (size_t)PLANE;     // NSIG * WSTRIDE floats
    unsigned int* ssqmax = (unsigned int*)(wts + NSIG * WSTRIDE);
    float* inv2g2 = (float*)(ssqmax + 1);

    static const int   Rs[NSIG] = {4, 12, 20, 28, 36};
    static const float Sg[NSIG] = {1.f, 3.f, 5.f, 7.f, 9.f};

    frangi_init_stats<<<1, 32, 0, stream>>>(ssqmax);
    frangi_weights<<<1, 128, 0, stream>>>(wts);
    frangi_gray<<<PLANE / 4 / 256, 256, 0, stream>>>(img, gray);

    const dim3 rowGrid(IMG_W / 256, IMG_H);
    const dim3 colGrid(8, 128);   // 8 waves/block, 16x32 strip per wave

    // sigma = 1: row pass, then gamma reduction pass
    frangi_rowconv<<<rowGrid, 256, 0, stream>>>(gray, wts, t0, t1, t2, Rs[0], 2 * Rs[0] + 1);
    frangi_colconv_wmma<<<colGrid, 256, 0, stream>>>(t0, t1, t2, wts, out, ssqmax,
                                                     inv2g2, Sg[0] * Sg[0],
                                                     Rs[0], 2 * Rs[0] + 1, 0);
    frangi_finalize<<<1, 1, 0, stream>>>(ssqmax, inv2g2);
    // sigma = 1: vals pass (stores, since d_out starts poisoned)
    frangi_colconv_wmma<<<colGrid, 256, 0, stream>>>(t0, t1, t2, wts, out, ssqmax,
                                                     inv2g2, Sg[0] * Sg[0],
                                                     Rs[0], 2 * Rs[0] + 1, 1);
    // sigma = 3,5,7,9: row pass + fused vals/max pass
    for (int s = 1; s < NSIG; ++s) {
        const float* wsig = wts + s * WSTRIDE;
        frangi_rowconv<<<rowGrid, 256, 0, stream>>>(gray, wsig, t0, t1, t2,
                                                    Rs[s], 2 * Rs[s] + 1);
        frangi_colconv_wmma<<<colGrid, 256, 0, stream>>>(t0, t1, t2, wsig, out, ssqmax,
                                                         inv2g2, Sg[s] * Sg[s],
                                                         Rs[s], 2 * Rs[s] + 1, 2);
    }
}